// _HAN_5729486373120
// MI455X (gfx1250) — compile-verified
//
#include <hip/hip_runtime.h>
#include <hip/hip_bf16.h>

// ---------------- problem constants (match reference) ----------------
#define Nn        50000
#define IN_DIM    256
#define NH        8        // heads
#define Dh        64       // per-head dim
#define HD        512      // NH*Dh
#define OUT_DIM   64
#define MP        2        // meta-paths
#define Ee        800000   // edges per metapath
#define SEM_HID   128
#define NEG_SLOPE 0.2f

typedef __attribute__((ext_vector_type(2))) float v2f;
typedef __attribute__((ext_vector_type(8))) float v8f;

// ---------------- gfx1250 async copy: global -> LDS, 16B per lane --------
__device__ __forceinline__ void async_b128(unsigned lds_addr, const void* gaddr) {
  asm volatile("global_load_async_to_lds_b128 %0, %1, off"
               :: "v"(lds_addr), "v"(gaddr) : "memory");
}
#define WAIT_ASYNCCNT(imm) asm volatile("s_wait_asynccnt " #imm ::: "memory")

// LDS byte offset of a __shared__ object (generic ptr low 32 bits = LDS offset)
__device__ __forceinline__ unsigned lds_off(const void* p) {
  return (unsigned)(unsigned long long)p;
}

// ------- order-preserving float<->uint key for atomicMax on floats -------
__device__ __forceinline__ unsigned ordkey(float f) {
  unsigned u = __float_as_uint(f);
  return (u & 0x80000000u) ? ~u : (u | 0x80000000u);
}
__device__ __forceinline__ float unordkey(unsigned k) {
  unsigned u = (k & 0x80000000u) ? (k & 0x7FFFFFFFu) : ~k;
  return __uint_as_float(u);
}

// ---------------- generic zero ----------------
__global__ void zero_b32(unsigned* __restrict__ p, int n) {
  int i = blockIdx.x * blockDim.x + threadIdx.x;
  if (i < n) p[i] = 0u;
}

// ---------------- WMMA fp32 GEMM with async-LDS double buffering ----------
// C = A @ B (+bias).  A:[nrows,K] row-major, B:[K,ncols] row-major.
// If FUSE_A: A-operand = betas[0]*A + betas[1]*A2, combined at fragment read.
// Block = 256 thr = 8 waves (2x4); block tile 32 rows x 64 cols; K chunk 32.
// Tiles staged in LDS via global_load_async_to_lds_b128 (ASYNCcnt),
// double-buffered: s_wait_asynccnt N overlaps next-chunk DMA with WMMAs.
template <bool FUSE_A, bool BIAS>
__global__ __launch_bounds__(256) void wmma_gemm(
    const float* __restrict__ A, const float* __restrict__ A2,
    const float* __restrict__ betas,
    const float* __restrict__ B, const float* __restrict__ bias,
    float* __restrict__ C, int nrows, int K, int ncols)
{
  constexpr int ANB       = FUSE_A ? 2 : 1;
  constexpr int AS_STRIDE = 36;               // padded: conflict-free + 16B rows
  constexpr int AS_TILE   = 32 * AS_STRIDE;   // floats per A matrix tile
  constexpr int BS_STRIDE = 72;               // padded: half-waves on disjoint banks
  constexpr int BS_TILE   = 32 * BS_STRIDE;

  __shared__ float As[2 * ANB * AS_TILE];
  __shared__ float Bs[2 * BS_TILE];

  const int tid     = threadIdx.x;
  const int lane    = tid & 31;
  const int wave    = tid >> 5;               // 0..7
  const int rowBase = blockIdx.x * 32;
  const int colBase = blockIdx.y * 64;

  // ---- cooperative tile-load mapping (per lane: one 16B quad) ----
  const int a_row = tid >> 3;                 // 0..31
  const int a_kq  = tid & 7;                  // quad of 4 floats along K
  int a_grow = rowBase + a_row;
  if (a_grow >= nrows) a_grow = nrows - 1;    // clamp tail rows (never stored)
  const unsigned a_lds = (unsigned)(a_row * AS_STRIDE + a_kq * 4) * 4u;

  const unsigned asBase = lds_off(&As[0]);
  const unsigned bsBase = lds_off(&Bs[0]);

  // ---- WMMA fragment mapping ----
  const int half  = lane >> 4;                // K-pair select
  const int l16   = lane & 15;
  const int arowL = (wave >> 2) * 16 + l16;   // local A row for fragment
  const int bcolL = (wave & 3) * 16 + l16;    // local B col for fragment

  float be0 = 0.f, be1 = 0.f;
  if (FUSE_A) { be0 = betas[0]; be1 = betas[1]; }

  const int nchunk = K / 32;
  v8f acc = {};

  // issue one K-chunk's async loads (per wave: ANB + 2 instructions)
  auto issueChunk = [&](int k0, int buf) {
    const unsigned aBuf = asBase + (unsigned)(buf * ANB * AS_TILE) * 4u;
    async_b128(aBuf + a_lds, A + (size_t)a_grow * K + k0 + a_kq * 4);
    if (FUSE_A)
      async_b128(aBuf + (unsigned)AS_TILE * 4u + a_lds,
                 A2 + (size_t)a_grow * K + k0 + a_kq * 4);
    const unsigned bBuf = bsBase + (unsigned)(buf * BS_TILE) * 4u;
#pragma unroll
    for (int j = 0; j < 2; ++j) {
      const int e  = tid * 4 + j * 1024;      // element in 32x64 tile
      const int kk = e >> 6, col = e & 63;    // col multiple of 4 -> 16B aligned
      async_b128(bBuf + (unsigned)(kk * BS_STRIDE + col) * 4u,
                 B + (size_t)(k0 + kk) * ncols + colBase + col);
    }
  };

  issueChunk(0, 0);
  for (int c = 0; c < nchunk; ++c) {
    if (c + 1 < nchunk) {
      issueChunk((c + 1) * 32, (c + 1) & 1);
      if (FUSE_A) { WAIT_ASYNCCNT(0x4); } else { WAIT_ASYNCCNT(0x3); }
    } else {
      WAIT_ASYNCCNT(0x0);
    }
    __syncthreads();                          // chunk-c tiles visible to all waves

    const float* Ab = &As[(c & 1) * ANB * AS_TILE];
    const float* Bb = &Bs[(c & 1) * BS_TILE];
#pragma unroll
    for (int kk = 0; kk < 32; kk += 4) {
      v2f a;
      const float* ap = Ab + arowL * AS_STRIDE + kk + half * 2;
      a.x = ap[0];
      a.y = ap[1];
      if (FUSE_A) {
        const float* ap2 = ap + AS_TILE;
        a.x = be0 * a.x + be1 * ap2[0];
        a.y = be0 * a.y + be1 * ap2[1];
      }
      v2f b;
      const float* bp = Bb + (kk + half * 2) * BS_STRIDE + bcolL;
      b.x = bp[0];
      b.y = bp[BS_STRIDE];
      acc = __builtin_amdgcn_wmma_f32_16x16x4_f32(
          false, a, false, b, (short)0, acc, false, false);
    }
    __syncthreads();                          // release buffer before overwrite
  }

  const int crow0 = rowBase + (wave >> 2) * 16;
  const int bcol  = colBase + bcolL;
  const float bb  = BIAS ? bias[bcol] : 0.0f;
#pragma unroll
  for (int v = 0; v < 8; ++v) {
    const int r = crow0 + v + half * 8;       // C/D layout: M = vgpr (+8 upper half)
    if (r < nrows) C[(size_t)r * ncols + bcol] = acc[v] + bb;
  }
}

// ---------------- el/er: per-node attention logits ----------------
__global__ void compute_lr(const float* __restrict__ feat,
                           const float* __restrict__ al, const float* __restrict__ ar,
                           float* __restrict__ el, float* __restrict__ er)
{
  int i = blockIdx.x * blockDim.x + threadIdx.x;   // i = n*NH + h
  if (i >= Nn * NH) return;
  int n = i >> 3, h = i & 7;
  const float4* f = (const float4*)(feat + (size_t)n * HD + h * Dh);
  const float4* a = (const float4*)(al + h * Dh);
  const float4* r = (const float4*)(ar + h * Dh);
  float sl = 0.f, sr = 0.f;
#pragma unroll
  for (int j = 0; j < Dh / 4; ++j) {
    float4 fv = f[j], av = a[j], rv = r[j];
    sl += fv.x * av.x + fv.y * av.y + fv.z * av.z + fv.w * av.w;
    sr += fv.x * rv.x + fv.y * rv.y + fv.z * rv.z + fv.w * rv.w;
  }
  el[i] = sl;
  er[i] = sr;
}

// ---------------- edge softmax pass 1: segment max over dst ----------------
__global__ void edge_max(const int* __restrict__ src, const int* __restrict__ dst,
                         const float* __restrict__ el, const float* __restrict__ er,
                         unsigned* __restrict__ mkey)
{
  int i = blockIdx.x * blockDim.x + threadIdx.x;   // i = e*NH + h
  if (i >= Ee * NH) return;
  int e = i >> 3, h = i & 7;
  int s = src[e], d = dst[e];
  float v = el[s * NH + h] + er[d * NH + h];
  v = v > 0.f ? v : NEG_SLOPE * v;                 // leaky_relu
  atomicMax(&mkey[d * NH + h], ordkey(v));
}

__global__ void m_decode(const unsigned* __restrict__ mkey, float* __restrict__ m)
{
  int i = blockIdx.x * blockDim.x + threadIdx.x;
  if (i >= Nn * NH) return;
  unsigned k = mkey[i];
  m[i] = (k == 0u) ? 0.0f : unordkey(k);           // untouched => zero in-degree => 0
}

// ---------------- edge softmax pass 2: sum of exp over dst ----------------
__global__ void edge_expsum(const int* __restrict__ src, const int* __restrict__ dst,
                            const float* __restrict__ el, const float* __restrict__ er,
                            const float* __restrict__ m, float* __restrict__ den)
{
  int i = blockIdx.x * blockDim.x + threadIdx.x;
  if (i >= Ee * NH) return;
  int e = i >> 3, h = i & 7;
  int s = src[e], d = dst[e];
  float v = el[s * NH + h] + er[d * NH + h];
  v = v > 0.f ? v : NEG_SLOPE * v;
  atomicAdd(&den[d * NH + h], expf(v - m[d * NH + h]));
}

// ---------------- weighted message aggregation (the bandwidth hog) --------
// One 256-thread block per edge; thread t covers feature elems 2t, 2t+1.
__global__ __launch_bounds__(256) void gat_aggregate(
    const int* __restrict__ src, const int* __restrict__ dst,
    const float* __restrict__ el, const float* __restrict__ er,
    const float* __restrict__ m, const float* __restrict__ den,
    const float* __restrict__ feat, float* __restrict__ agg)
{
  int e = blockIdx.x;
  int s = src[e], d = dst[e];
  int idx = threadIdx.x * 2;                       // 0..510
  int h = idx >> 6;                                // Dh = 64
  float v = el[s * NH + h] + er[d * NH + h];
  v = v > 0.f ? v : NEG_SLOPE * v;
  float ex = expf(v - m[d * NH + h]);
  float alpha = ex / fmaxf(den[d * NH + h], 1e-9f);
  float2 f = *(const float2*)(feat + (size_t)s * HD + idx);
  atomicAdd(&agg[(size_t)d * HD + idx],     alpha * f.x);
  atomicAdd(&agg[(size_t)d * HD + idx + 1], alpha * f.y);
}

// ---------------- z = elu(agg + bg) in place ----------------
__global__ void bias_elu(float* __restrict__ z, const float* __restrict__ bg)
{
  int i = blockIdx.x * blockDim.x + threadIdx.x;
  if (i >= Nn * HD) return;
  float x = z[i] + bg[i & (HD - 1)];
  z[i] = x > 0.f ? x : expm1f(x);
}

// ---------------- semantic attention: wsum[p] += sum_n tanh(t[n,:]) . q ----
__global__ void sem_reduce(const float* __restrict__ t, const float* __restrict__ q,
                           float* __restrict__ wsum, int midx)
{
  __shared__ float red[256];
  int n = blockIdx.x * blockDim.x + threadIdx.x;
  float acc = 0.f;
  if (n < Nn) {
    const float* tr = t + (size_t)n * SEM_HID;
#pragma unroll 4
    for (int k = 0; k < SEM_HID; ++k) acc += tanhf(tr[k]) * q[k];
  }
  red[threadIdx.x] = acc;
  __syncthreads();
  for (int s = 128; s > 0; s >>= 1) {
    if (threadIdx.x < s) red[threadIdx.x] += red[threadIdx.x + s];
    __syncthreads();
  }
  if (threadIdx.x == 0) atomicAdd(&wsum[midx], red[0]);
}

__global__ void beta_softmax(const float* __restrict__ wsum, float* __restrict__ beta)
{
  if (threadIdx.x == 0) {
    float w0 = wsum[0] / (float)Nn, w1 = wsum[1] / (float)Nn;
    float mx = fmaxf(w0, w1);
    float e0 = expf(w0 - mx), e1 = expf(w1 - mx);
    float s = e0 + e1;
    beta[0] = e0 / s;
    beta[1] = e1 / s;
  }
}

// =====================================================================
extern "C" void kernel_launch(void* const* d_in, const int* in_sizes, int n_in,
                              void* d_out, int out_size, void* d_ws, size_t ws_size,
                              hipStream_t stream)
{
  (void)in_sizes; (void)n_in; (void)out_size; (void)ws_size;

  const float* x      = (const float*)d_in[0];
  const int*   srcAll = (const int*)  d_in[1];
  const int*   dstAll = (const int*)  d_in[2];
  const float* Wg     = (const float*)d_in[3];
  const float* al     = (const float*)d_in[4];
  const float* ar     = (const float*)d_in[5];
  const float* bg     = (const float*)d_in[6];
  const float* W1     = (const float*)d_in[7];
  const float* b1     = (const float*)d_in[8];
  const float* q      = (const float*)d_in[9];
  const float* Wout   = (const float*)d_in[10];
  const float* bout   = (const float*)d_in[11];

  // ---------- workspace carve-up (256B aligned) ----------
  char*  ws  = (char*)d_ws;
  size_t off = 0;
  auto carve = [&](size_t bytes) -> void* {
    void* p = ws + off;
    off = (off + bytes + 255) & ~(size_t)255;
    return p;
  };
  float*    feat = (float*)   carve((size_t)Nn * HD * 4);        // per-metapath GAT features
  float*    z0   = (float*)   carve((size_t)Nn * HD * 4);        // metapath-0 output
  float*    z1   = (float*)   carve((size_t)Nn * HD * 4);        // metapath-1 output
  float*    tbuf = (float*)   carve((size_t)Nn * SEM_HID * 4);   // tanh-projection scratch
  float*    el   = (float*)   carve((size_t)Nn * NH * 4);
  float*    er   = (float*)   carve((size_t)Nn * NH * 4);
  float*    mmax = (float*)   carve((size_t)Nn * NH * 4);
  float*    den  = (float*)   carve((size_t)Nn * NH * 4);
  unsigned* mkey = (unsigned*)carve((size_t)Nn * NH * 4);
  float*    wsum = (float*)   carve(256);
  float*    beta = (float*)   carve(256);

  auto cdiv = [](int a, int b) { return (a + b - 1) / b; };
  const int gRowBlk = cdiv(Nn, 32);   // 1563

  zero_b32<<<1, 256, 0, stream>>>((unsigned*)wsum, 2);

  for (int p = 0; p < MP; ++p) {
    float* z = p ? z1 : z0;
    const int* src = srcAll + (size_t)p * Ee;
    const int* dst = dstAll + (size_t)p * Ee;

    // reset accumulators (replay-safe: everything re-zeroed each call)
    zero_b32<<<cdiv(Nn * HD, 256), 256, 0, stream>>>((unsigned*)z, Nn * HD);
    zero_b32<<<cdiv(Nn * NH, 256), 256, 0, stream>>>(mkey, Nn * NH);
    zero_b32<<<cdiv(Nn * NH, 256), 256, 0, stream>>>((unsigned*)den, Nn * NH);

    // feat = x @ Wg[p]   [50000,256]x[256,512]  (WMMA fp32, async-LDS staged)
    wmma_gemm<false, false><<<dim3(gRowBlk, HD / 64), 256, 0, stream>>>(
        x, nullptr, nullptr, Wg + (size_t)p * IN_DIM * HD, nullptr,
        feat, Nn, IN_DIM, HD);

    compute_lr<<<cdiv(Nn * NH, 256), 256, 0, stream>>>(
        feat, al + (size_t)p * NH * Dh, ar + (size_t)p * NH * Dh, el, er);

    // edge softmax (max / exp-sum), then weighted scatter-aggregation
    edge_max   <<<cdiv(Ee * NH, 256), 256, 0, stream>>>(src, dst, el, er, mkey);
    m_decode   <<<cdiv(Nn * NH, 256), 256, 0, stream>>>(mkey, mmax);
    edge_expsum<<<cdiv(Ee * NH, 256), 256, 0, stream>>>(src, dst, el, er, mmax, den);
    gat_aggregate<<<Ee, 256, 0, stream>>>(src, dst, el, er, mmax, den, feat, z);

    bias_elu<<<cdiv(Nn * HD, 256), 256, 0, stream>>>(z, bg + (size_t)p * HD);

    // semantic attention projection: t = z @ W1 + b1  (WMMA fp32)
    wmma_gemm<false, true><<<dim3(gRowBlk, SEM_HID / 64), 256, 0, stream>>>(
        z, nullptr, nullptr, W1, b1, tbuf, Nn, HD, SEM_HID);
    sem_reduce<<<cdiv(Nn, 256), 256, 0, stream>>>(tbuf, q, wsum, p);
  }

  beta_softmax<<<1, 32, 0, stream>>>(wsum, beta);

  // out = (beta0*z0 + beta1*z1) @ Wout + bout  (fused-A WMMA fp32)
  wmma_gemm<true, true><<<dim3(gRowBlk, OUT_DIM / 64), 256, 0, stream>>>(
      z0, z1, beta, Wout, bout, (float*)d_out, Nn, HD, OUT_DIM);
}